// JointNetwork_14216341749903
// MI455X (gfx1250) — compile-verified
//
#include <hip/hip_runtime.h>
#include <hip/hip_bf16.h>

typedef float v2f __attribute__((ext_vector_type(2)));
typedef float v4f __attribute__((ext_vector_type(4)));
typedef float v8f __attribute__((ext_vector_type(8)));

// -----------------------------------------------------------------------------
// c[v] = sum_h b0[h] * W1[h,v] + b1[v]      (tiny: 0.5 MFLOP)
// -----------------------------------------------------------------------------
__global__ __launch_bounds__(256) void bias_combine_kernel(
    const float* __restrict__ b0, const float* __restrict__ W1,
    const float* __restrict__ b1, float* __restrict__ cvec,
    int Hdim, int Vdim)
{
    int v = blockIdx.x * blockDim.x + threadIdx.x;
    if (v >= Vdim) return;
    float s = b1[v];
    for (int h = 0; h < Hdim; ++h)
        s += b0[h] * W1[(size_t)h * Vdim + v];
    cvec[v] = s;
}

// -----------------------------------------------------------------------------
// C[M,N] = A[M,K] @ B[K,N] (+ bias[N])  row-major fp32, via V_WMMA_F32_16X16X4_F32
// Requires: M%64==0, N%64==0, K%4==0.  Block = 128 threads (4 waves).
// Each wave computes a 16(M) x 64(N) strip; block covers 64x64.
//
// Fragment layouts (ISA 7.12.2, wave32):
//   A 16x4:  lanes 0-15 -> row M=lane, v.{x,y} = K = {0,1};  lanes 16-31 -> K={2,3}
//   B 4x16:  lanes 0-15 -> col N=lane, v.{x,y} = K = {0,1};  lanes 16-31 -> K={2,3}
//   C 16x16: lane%16 = N, VGPR r -> M = r + 8*(lane/16)
// -----------------------------------------------------------------------------
__global__ __launch_bounds__(128) void gemm_f32_wmma_kernel(
    const float* __restrict__ A, const float* __restrict__ B,
    const float* __restrict__ bias, float* __restrict__ C,
    int Ndim, int Kdim)
{
    const int lane = threadIdx.x & 31;
    const int wave = threadIdx.x >> 5;
    const int l    = lane & 15;
    const int half = lane >> 4;

    const int n0 = blockIdx.x * 64;
    const int m0 = blockIdx.y * 64 + wave * 16;

    v8f acc0 = {}, acc1 = {}, acc2 = {}, acc3 = {};

    const float* Arow = A + (size_t)(m0 + l) * Kdim + 2 * half;

    #pragma unroll 2
    for (int k = 0; k < Kdim; k += 4) {
        // A fragment: 8-byte aligned pair (k even)
        v2f a = *(const v2f*)(Arow + k);

        const float* Bp0 = B + (size_t)(k + 2 * half) * Ndim + n0 + l;
        const float* Bp1 = Bp0 + Ndim;
        v2f b0v; b0v.x = Bp0[0];  b0v.y = Bp1[0];
        v2f b1v; b1v.x = Bp0[16]; b1v.y = Bp1[16];
        v2f b2v; b2v.x = Bp0[32]; b2v.y = Bp1[32];
        v2f b3v; b3v.x = Bp0[48]; b3v.y = Bp1[48];

        acc0 = __builtin_amdgcn_wmma_f32_16x16x4_f32(false, a, false, b0v, (short)0, acc0, false, false);
        acc1 = __builtin_amdgcn_wmma_f32_16x16x4_f32(false, a, false, b1v, (short)0, acc1, false, false);
        acc2 = __builtin_amdgcn_wmma_f32_16x16x4_f32(false, a, false, b2v, (short)0, acc2, false, false);
        acc3 = __builtin_amdgcn_wmma_f32_16x16x4_f32(false, a, false, b3v, (short)0, acc3, false, false);
    }

    float bv0 = 0.f, bv1 = 0.f, bv2 = 0.f, bv3 = 0.f;
    if (bias) {
        bv0 = bias[n0 + l];
        bv1 = bias[n0 + 16 + l];
        bv2 = bias[n0 + 32 + l];
        bv3 = bias[n0 + 48 + l];
    }

    #pragma unroll
    for (int r = 0; r < 8; ++r) {
        int m = m0 + r + 8 * half;
        float* Crow = C + (size_t)m * Ndim + n0 + l;
        Crow[0]  = acc0[r] + bv0;
        Crow[16] = acc1[r] + bv1;
        Crow[32] = acc2[r] + bv2;
        Crow[48] = acc3[r] + bv3;
    }
}

// -----------------------------------------------------------------------------
// out[b,t,u,v] = EE[b*T+t, v] + PP[b*U+u, v]
// 512 MB streaming store -> non-temporal b128; EE/PP rows are L2-resident (5 MB).
// One block per (b,t,u) row; 256 threads x v4f = V=1024 floats.
// -----------------------------------------------------------------------------
__global__ __launch_bounds__(256) void epilogue_add_kernel(
    const float* __restrict__ EE, const float* __restrict__ PP,
    float* __restrict__ out, int T, int U, int Vdim)
{
    const int idx = blockIdx.x;        // b*T*U + t*U + u
    const int u   = idx % U;
    const int bt  = idx / U;           // b*T + t
    const int b   = bt / T;

    const v4f* e = (const v4f*)(EE + (size_t)bt * Vdim);
    const v4f* p = (const v4f*)(PP + (size_t)(b * U + u) * Vdim);
    v4f*       o = (v4f*)(out + (size_t)idx * Vdim);

    const int i = threadIdx.x;         // Vdim/4 == blockDim.x
    v4f r = e[i] + p[i];
    __builtin_nontemporal_store(r, &o[i]);
}

// -----------------------------------------------------------------------------
extern "C" void kernel_launch(void* const* d_in, const int* in_sizes, int n_in,
                              void* d_out, int out_size, void* d_ws, size_t ws_size,
                              hipStream_t stream) {
    (void)in_sizes; (void)n_in; (void)out_size; (void)ws_size;

    constexpr int B = 2, T = 512, U = 128, D = 512, H = 512, V = 1024;

    const float* pred = (const float*)d_in[0];  // [B,U,D]
    const float* enc  = (const float*)d_in[1];  // [B,T,D]
    const float* W0   = (const float*)d_in[2];  // [D,H]
    const float* b0   = (const float*)d_in[3];  // [H]
    const float* W1   = (const float*)d_in[4];  // [H,V]
    const float* b1   = (const float*)d_in[5];  // [V]
    float* out = (float*)d_out;                 // [B,T,U,V]

    float* ws   = (float*)d_ws;
    float* Mmat = ws;                           // [D,V]   2 MB
    float* cvec = Mmat + (size_t)D * V;         // [V]     4 KB
    float* EE   = cvec + V;                     // [B*T,V] 4 MB
    float* PP   = EE + (size_t)B * T * V;       // [B*U,V] 1 MB

    // c = b0 @ W1 + b1
    bias_combine_kernel<<<(V + 255) / 256, 256, 0, stream>>>(b0, W1, b1, cvec, H, V);

    // M = W0 @ W1                     (512 x 1024, K=512)
    gemm_f32_wmma_kernel<<<dim3(V / 64, D / 64), 128, 0, stream>>>(
        W0, W1, nullptr, Mmat, V, H);

    // EE = enc @ M + c                (1024 x 1024, K=512)
    gemm_f32_wmma_kernel<<<dim3(V / 64, (B * T) / 64), 128, 0, stream>>>(
        enc, Mmat, cvec, EE, V, D);

    // PP = pred @ M                   (256 x 1024, K=512)
    gemm_f32_wmma_kernel<<<dim3(V / 64, (B * U) / 64), 128, 0, stream>>>(
        pred, Mmat, nullptr, PP, V, D);

    // out[b,t,u,:] = EE[b,t,:] + PP[b,u,:]   (512 MB NT stream)
    epilogue_add_kernel<<<B * T * U, 256, 0, stream>>>(EE, PP, out, T, U, V);
}